// Pixelcoreg_Focalloss_52312701665541
// MI455X (gfx1250) — compile-verified
//
#include <hip/hip_runtime.h>
#include <stdint.h>

#define BATCH    8
#define HW_      (512*512)
#define NTOT     (BATCH*HW_)
#define HW_SHIFT 18

// control block layout (uint32 words), placed after keys[] in workspace
#define C_K      0    // num_remember
#define C_TOTT   1    // total targets sum (int)
#define C_SUML3  2    // selected loss_3 sum (float bits, atomic)
#define C_SELT   3    // selected targets count (int)
#define C_PREFIX 4    // [8] per-row radix prefix -> final threshold key
#define C_KREM   12   // [8] per-row remaining rank -> final #ties to take
#define C_EQ     20   // [8] per-row tie counters
#define C_HIST   32   // [8][256] histograms
#define C_WORDS  (32 + 8*256)

typedef __attribute__((ext_vector_type(2))) float v2f;
typedef __attribute__((ext_vector_type(8))) float v8f;

// monotone float -> uint mapping (ascending order preserved)
__device__ __forceinline__ uint32_t f2ord(float f) {
    uint32_t u = __float_as_uint(f);
    return (u & 0x80000000u) ? ~u : (u | 0x80000000u);
}

// two-class log-softmax + focal term, matching the reference's exact formula
__device__ __forceinline__ void head_eval(float x0, float x1, float tf,
                                          float& p0, float& p1,
                                          float& lp0, float& lp1, float& l) {
    float m  = fmaxf(x0, x1);
    float e0 = expf(x0 - m);
    float e1 = expf(x1 - m);
    float ls = logf(e0 + e1);
    lp0 = (x0 - m) - ls;
    lp1 = (x1 - m) - ls;
    p0  = expf(lp0);            // p = exp(logp), as in the reference
    p1  = expf(lp1);
    float q1 = 1.0f - p1;
    float q0 = 1.0f - p0;
    l = -tf * q1 * q1 * lp1 - (1.0f - tf) * q0 * q0 * lp0;
}

__global__ void init_kernel(const float* __restrict__ fr_p, uint32_t* __restrict__ ctrl) {
    int tid = threadIdx.x;
    float fr = *fr_p;
    int k = (int)((1.0f - fr) * (float)HW_);   // int() truncation like Python
    if (tid == 0) {
        ctrl[C_K] = (uint32_t)k;
        ctrl[C_TOTT] = 0u; ctrl[C_SUML3] = 0u; ctrl[C_SELT] = 0u;
    }
    if (tid < 8) {
        ctrl[C_PREFIX + tid] = 0u;
        ctrl[C_KREM + tid]   = (uint32_t)k;
        ctrl[C_EQ + tid]     = 0u;
    }
    for (int j = 0; j < 8; ++j) ctrl[C_HIST + j * 256 + tid] = 0u;
}

// 4 pixels/thread, b128 loads; 2048 blocks x 256 threads covers NTOT exactly
__global__ void compute_kernel(const float* __restrict__ in1, const float* __restrict__ in2,
                               const float* __restrict__ in3, const int* __restrict__ tgt,
                               const float* __restrict__ kdw_p,
                               uint32_t* __restrict__ keys, uint32_t* __restrict__ ctrl) {
    int tid = threadIdx.x;
    int i4  = (blockIdx.x * 256 + tid) * 4;
    int b   = i4 >> HW_SHIFT;                 // block spans 1024 px, all in one row
    int p   = i4 & (HW_ - 1);
    int base = b * 2 * HW_ + p;
    __builtin_prefetch(in3 + base + 8192, 0, 0);   // -> global_prefetch_b8

    float4 a0 = *(const float4*)(in1 + base);
    float4 a1 = *(const float4*)(in1 + base + HW_);
    float4 b0 = *(const float4*)(in2 + base);
    float4 b1 = *(const float4*)(in2 + base + HW_);
    float4 c0 = *(const float4*)(in3 + base);
    float4 c1 = *(const float4*)(in3 + base + HW_);
    int4   t4 = *(const int4*)(tgt + i4);
    float kdw = *kdw_p;

    uint32_t kout[4];
    int tcnt = 0;
#pragma unroll
    for (int j = 0; j < 4; ++j) {
        int   t  = (&t4.x)[j];
        float tf = (float)t;
        tcnt += (t != 0);
        float p0a, p1a, lp0a, lp1a, la;
        head_eval((&a0.x)[j], (&a1.x)[j], tf, p0a, p1a, lp0a, lp1a, la);
        float p0b, p1b, lp0b, lp1b, lb;
        head_eval((&b0.x)[j], (&b1.x)[j], tf, p0b, p1b, lp0b, lp1b, lb);
        float p0c, p1c, lp0c, lp1c, lc;
        head_eval((&c0.x)[j], (&c1.x)[j], tf, p0c, p1c, lp0c, lp1c, lc);
        float kd12 = p0a * (lp0a - lp0b) + p1a * (lp1a - lp1b);
        float kd21 = p0b * (lp0b - lp0a) + p1b * (lp1b - lp1a);
        float loss = (1.0f - kdw) * (la + lb + lc) + kdw * (kd12 + kd21);
        kout[j] = f2ord(loss);
    }
    *(uint4*)(keys + i4) = make_uint4(kout[0], kout[1], kout[2], kout[3]);

    // exact integer targets.sum()
    __shared__ int sT;
    if (tid == 0) sT = 0;
    __syncthreads();
    if (tcnt) atomicAdd(&sT, tcnt);
    __syncthreads();
    if (tid == 0 && sT) atomicAdd((int*)&ctrl[C_TOTT], sT);
}

// one radix-select byte pass; 32 blocks per row, 8192 keys per block, b128 loads
__global__ void hist_kernel(const uint32_t* __restrict__ keys, uint32_t* __restrict__ ctrl,
                            int shift_b, int hi_shift) {
    __shared__ uint32_t h[256];
    int tid = threadIdx.x;
    h[tid] = 0u;
    int row   = blockIdx.x >> 5;
    int chunk = blockIdx.x & 31;
    const uint4* kp4 = (const uint4*)(keys + row * HW_ + chunk * 8192);
    uint32_t pref = ctrl[C_PREFIX + row];
    __syncthreads();
    for (int it = 0; it < 8; ++it) {
        uint4 kv = kp4[it * 256 + tid];
#pragma unroll
        for (int j = 0; j < 4; ++j) {
            uint32_t key = (&kv.x)[j];
            if ((uint32_t)(((uint64_t)key) >> hi_shift) == pref)
                atomicAdd(&h[(key >> shift_b) & 255u], 1u);
        }
    }
    __syncthreads();
    uint32_t c = h[tid];
    if (c) atomicAdd(&ctrl[C_HIST + row * 256 + tid], c);
}

// pick the bin holding the kr-th element per row, extend prefix, re-zero hist
__global__ void scan_kernel(uint32_t* __restrict__ ctrl) {
    int tid = threadIdx.x;
    if (tid < 8) {
        uint32_t kr  = ctrl[C_KREM + tid];
        uint32_t* h  = &ctrl[C_HIST + tid * 256];
        uint32_t cum = 0, bv = 255u;
        for (int bno = 0; bno < 256; ++bno) {
            uint32_t c = h[bno];
            if (cum + c >= kr) { bv = (uint32_t)bno; kr = kr - cum; break; }
            cum += c;
        }
        ctrl[C_PREFIX + tid] = (ctrl[C_PREFIX + tid] << 8) | bv;
        ctrl[C_KREM + tid]   = kr;
    }
    __syncthreads();
    for (int j = 0; j < 8; ++j) ctrl[C_HIST + j * 256 + tid] = 0u;
}

// 4 pixels/thread; threshold-select, then wave sum on the matrix pipe
__global__ void final_kernel(const float* __restrict__ in3, const int* __restrict__ tgt,
                             const uint32_t* __restrict__ keys, uint32_t* __restrict__ ctrl) {
    int tid = threadIdx.x;
    int i4  = (blockIdx.x * 256 + tid) * 4;
    int row = i4 >> HW_SHIFT;
    int p   = i4 & (HW_ - 1);
    int base = row * 2 * HW_ + p;

    float4 c0 = *(const float4*)(in3 + base);
    float4 c1 = *(const float4*)(in3 + base + HW_);
    int4   t4 = *(const int4*)(tgt + i4);
    uint4  kv = *(const uint4*)(keys + i4);
    uint32_t T  = ctrl[C_PREFIX + row];
    uint32_t ne = ctrl[C_KREM + row];

    float val = 0.0f;
    int   cnt = 0;
#pragma unroll
    for (int j = 0; j < 4; ++j) {
        int   t  = (&t4.x)[j];
        float tf = (float)t;
        float p0, p1, lp0, lp1, lc;
        head_eval((&c0.x)[j], (&c1.x)[j], tf, p0, p1, lp0, lp1, lc);
        uint32_t key = (&kv.x)[j];
        bool inc = key < T;
        if (key == T) {                    // tie handling: take exactly kRemain ties
            uint32_t old = atomicAdd(&ctrl[C_EQ + row], 1u);
            inc = old < ne;
        }
        if (inc) { val += lc; cnt += (t != 0); }
    }

    // wave-level sum via matrix pipe: ones(16x4) x B -> 16 column sums that
    // partition all 64 B cells, so exact regardless of B's (K,N) lane mapping.
    float wsum;
#if __has_builtin(__builtin_amdgcn_wmma_f32_16x16x4_f32)
    v2f amat; amat[0] = 1.0f; amat[1] = 1.0f;
    v2f bmat; bmat[0] = val;  bmat[1] = 0.0f;
    v8f acc = {};
    acc = __builtin_amdgcn_wmma_f32_16x16x4_f32(
        false, amat, false, bmat, (short)0, acc, false, false);
    wsum = acc[0];                         // D[0][n] / D[8][n]: identical colsums
    wsum += __shfl_xor(wsum, 1, 16);
    wsum += __shfl_xor(wsum, 2, 16);
    wsum += __shfl_xor(wsum, 4, 16);
    wsum += __shfl_xor(wsum, 8, 16);
#else
    wsum = val;
    for (int off = 16; off > 0; off >>= 1) wsum += __shfl_xor(wsum, off, 32);
#endif

    __shared__ float wl[8];
    __shared__ int   sC;
    if (tid == 0) sC = 0;
    __syncthreads();
    int wid = tid >> 5;
    if ((tid & 31) == 0) wl[wid] = wsum;
    if (cnt) atomicAdd(&sC, cnt);
    __syncthreads();
    if (tid == 0) {
        float s = 0.0f;
        for (int w = 0; w < 8; ++w) s += wl[w];
        atomicAdd((float*)&ctrl[C_SUML3], s);
        if (sC) atomicAdd((int*)&ctrl[C_SELT], sC);
    }
}

__global__ void finalize_kernel(const uint32_t* __restrict__ ctrl, float* __restrict__ out) {
    float s3  = __uint_as_float(ctrl[C_SUML3]);
    int   k   = (int)ctrl[C_K];
    int   sel = (int)ctrl[C_SELT];
    int   tot = (int)ctrl[C_TOTT];
    out[0] = s3 / (float)(BATCH * k);
    out[1] = (float)sel / (float)tot;
}

extern "C" void kernel_launch(void* const* d_in, const int* in_sizes, int n_in,
                              void* d_out, int out_size, void* d_ws, size_t ws_size,
                              hipStream_t stream) {
    (void)in_sizes; (void)n_in; (void)out_size; (void)ws_size;
    const float* in1 = (const float*)d_in[0];
    const float* in2 = (const float*)d_in[1];
    const float* in3 = (const float*)d_in[2];
    const int*   tgt = (const int*)d_in[3];
    const float* fr  = (const float*)d_in[4];
    const float* kdw = (const float*)d_in[5];
    float* out = (float*)d_out;

    uint32_t* keys = (uint32_t*)d_ws;          // NTOT words (8 MB)
    uint32_t* ctrl = keys + NTOT;              // C_WORDS words

    init_kernel<<<1, 256, 0, stream>>>(fr, ctrl);
    compute_kernel<<<NTOT / 1024, 256, 0, stream>>>(in1, in2, in3, tgt, kdw, keys, ctrl);
    hist_kernel<<<256, 256, 0, stream>>>(keys, ctrl, 24, 32);
    scan_kernel<<<1, 256, 0, stream>>>(ctrl);
    hist_kernel<<<256, 256, 0, stream>>>(keys, ctrl, 16, 24);
    scan_kernel<<<1, 256, 0, stream>>>(ctrl);
    hist_kernel<<<256, 256, 0, stream>>>(keys, ctrl, 8, 16);
    scan_kernel<<<1, 256, 0, stream>>>(ctrl);
    hist_kernel<<<256, 256, 0, stream>>>(keys, ctrl, 0, 8);
    scan_kernel<<<1, 256, 0, stream>>>(ctrl);
    final_kernel<<<NTOT / 1024, 256, 0, stream>>>(in3, tgt, keys, ctrl);
    finalize_kernel<<<1, 1, 0, stream>>>(ctrl, out);
}